// Model_47485158425103
// MI455X (gfx1250) — compile-verified
//
#include <hip/hip_runtime.h>

// ---------------------------------------------------------------------------
// CDNA5 / gfx1250 implementation of: SAGEConv(sum, root_weight=False) ->
// ReLU -> global_mean_pool -> Linear -> ReLU.
//
// Workspace layout (floats):  agg[N] | sums[G*H] | counts[G]
// ---------------------------------------------------------------------------

typedef __attribute__((ext_vector_type(2))) float v2f;
typedef __attribute__((ext_vector_type(8))) float v8f;
typedef __attribute__((ext_vector_type(4))) int v4i;

#define NPB 256  // nodes per block in the fused WMMA pooling kernel

// ---------------------------------------------------------------------------
__global__ void zero_ws_kernel(float* __restrict__ ws, int total) {
  int i = blockIdx.x * blockDim.x + threadIdx.x;
  if (i < total) ws[i] = 0.0f;
}

// ---------------------------------------------------------------------------
// agg[dst] += x[src] over all edges. x/agg are ~400KB each -> L2 resident;
// the 51MB index stream dominates (HBM), so index loads are b128 and
// non-temporal to avoid displacing the resident gather/scatter working set.
// ---------------------------------------------------------------------------
__global__ void edge_scatter_vec4_kernel(const v4i* __restrict__ src4,
                                         const v4i* __restrict__ dst4,
                                         const float* __restrict__ x,
                                         float* __restrict__ agg, int E4) {
  int i = blockIdx.x * blockDim.x + threadIdx.x;
  if (i >= E4) return;
  v4i s = __builtin_nontemporal_load(&src4[i]);
  v4i d = __builtin_nontemporal_load(&dst4[i]);
  atomicAdd(&agg[d.x], x[s.x]);
  atomicAdd(&agg[d.y], x[s.y]);
  atomicAdd(&agg[d.z], x[s.z]);
  atomicAdd(&agg[d.w], x[s.w]);
}

__global__ void edge_scatter_scalar_kernel(const int* __restrict__ src,
                                           const int* __restrict__ dst,
                                           const float* __restrict__ x,
                                           float* __restrict__ agg, int E) {
  int i = blockIdx.x * blockDim.x + threadIdx.x;
  int stride = gridDim.x * blockDim.x;
  for (; i < E; i += stride) atomicAdd(&agg[dst[i]], x[src[i]]);
}

// ---------------------------------------------------------------------------
// Fused: h = relu(agg * W_l + b_l) (rank-1, never materialized) and
// per-graph segment sum expressed as WMMA:
//   C[16 graph-slots x 16 cols] += A_onehot[16 x 4nodes] * B[4nodes x 16 cols]
// Block = 256 threads = 8 waves; wave w owns columns [32w, 32w+32) as two
// 16-wide WMMA tiles. The block's NPB nodes are staged into LDS with
// GLOBAL_LOAD_ASYNC_TO_LDS_B32 (ASYNCcnt) and padded with a sentinel batch
// id so the hot loop needs zero validity checks.
// ---------------------------------------------------------------------------
__global__ void __launch_bounds__(256)
node_pool_wmma_kernel(const float* __restrict__ agg,
                      const int* __restrict__ batch,
                      const float* __restrict__ W_l,
                      const float* __restrict__ b_l,
                      float* __restrict__ sums,
                      float* __restrict__ counts,
                      int N, int H, int G) {
  __shared__ float s_agg[NPB];
  __shared__ int s_bat[NPB];

  const int nb = blockIdx.x * NPB;
  const int tid = threadIdx.x;  // blockDim.x == NPB
  const int gi = nb + tid;
  const bool valid = gi < N;

  if (valid) {
    // LDS byte offsets: flat pointers into the LDS aperture keep the LDS
    // offset in addr[31:0] (ISA 10.2), so truncation yields the DS address.
    unsigned lds_a = (unsigned)(uintptr_t)&s_agg[tid];
    unsigned lds_b = (unsigned)(uintptr_t)&s_bat[tid];
    unsigned go = (unsigned)gi * 4u;
    asm volatile("global_load_async_to_lds_b32 %0, %1, %2"
                 :
                 : "v"(lds_a), "v"(go), "s"(agg)
                 : "memory");
    asm volatile("global_load_async_to_lds_b32 %0, %1, %2"
                 :
                 : "v"(lds_b), "v"(go), "s"(batch)
                 : "memory");
  } else {
    s_agg[tid] = 0.0f;
    s_bat[tid] = (int)0xE0000000;  // sentinel: never matches any slot
  }
  asm volatile("s_wait_asynccnt 0x0" ::: "memory");
  __syncthreads();

  // node counts per graph (for the mean)
  if (valid) atomicAdd(&counts[s_bat[tid]], 1.0f);

  const int lane = tid & 31;
  const int wave = tid >> 5;
  const int n = lane & 15;     // tile column; also A's M row
  const int half = lane >> 4;  // 0 or 1
  const int c0 = wave * 32;

  // per-lane column params for the two 16-wide tiles
  const float wl0 = W_l[c0 + n];
  const float bl0 = b_l[c0 + n];
  const float wl1 = W_l[c0 + 16 + n];
  const float bl1 = b_l[c0 + 16 + n];

  // A layout (16x4 f32): lane holds A[M=n][K=kA], A[M=n][K=kA+1]
  const int kA = half * 2;
  // B layout (4x16 f32): lane holds B[K=kB][N=n], B[K=kB+2][N=n]
  const int kB = half;

  v8f acc0 = {};
  v8f acc1 = {};
  int gbase = s_bat[0];

  // flush one C tile: C[v][lane] = element (M = v + 8*half, N = n)
  auto flush = [&](v8f& a, int c) {
    const int mbase = half * 8;
#pragma unroll
    for (int v = 0; v < 8; ++v) {
      float val = a[v];
      int g = gbase + mbase + v;
      if (val != 0.0f && g < G) atomicAdd(&sums[g * H + c + n], val);
    }
  };

  int nvalid = N - nb;
  if (nvalid > NPB) nvalid = NPB;

  for (int k0 = 0; k0 < nvalid; k0 += 4) {
    int t0 = s_bat[k0 + 0];
    int t1 = s_bat[k0 + 1];
    int t2 = s_bat[k0 + 2];
    int t3 = s_bat[k0 + 3];
    int m01 = t0 > t1 ? t0 : t1;
    int m23 = t2 > t3 ? t2 : t3;
    int bmax = m01 > m23 ? m01 : m23;
    if (bmax - gbase > 15) {  // slot window overflow: flush + rebase
      flush(acc0, c0);
      flush(acc1, c0 + 16);
      acc0 = v8f{};
      acc1 = v8f{};
      gbase = t0;  // chunk-start node is always real (k0 < nvalid)
    }

    // one-hot membership for this lane's two K slots (sentinels never match)
    v2f A;
    A.x = (s_bat[k0 + kA] - gbase == n) ? 1.0f : 0.0f;
    A.y = (s_bat[k0 + kA + 1] - gbase == n) ? 1.0f : 0.0f;

    // B = relu(agg*w + b); padded nodes have zero A column, value irrelevant
    float ag0 = s_agg[k0 + kB];
    float ag1 = s_agg[k0 + kB + 2];

    v2f B0 = {fmaxf(fmaf(ag0, wl0, bl0), 0.0f),
              fmaxf(fmaf(ag1, wl0, bl0), 0.0f)};
    acc0 = __builtin_amdgcn_wmma_f32_16x16x4_f32(false, A, false, B0,
                                                 (short)0, acc0, false, false);

    v2f B1 = {fmaxf(fmaf(ag0, wl1, bl1), 0.0f),
              fmaxf(fmaf(ag1, wl1, bl1), 0.0f)};
    acc1 = __builtin_amdgcn_wmma_f32_16x16x4_f32(false, A, false, B1,
                                                 (short)0, acc1, false, false);
  }

  flush(acc0, c0);
  flush(acc1, c0 + 16);
}

// ---------------------------------------------------------------------------
// out[g] = relu( (sums[g,:]/max(cnt,1)) . W_out + b_out )
// ---------------------------------------------------------------------------
__global__ void finalize_kernel(const float* __restrict__ sums,
                                const float* __restrict__ counts,
                                const float* __restrict__ W_out,
                                const float* __restrict__ b_out,
                                float* __restrict__ out, int H) {
  __shared__ float red[256];
  int g = blockIdx.x;
  int t = threadIdx.x;
  float cnt = fmaxf(counts[g], 1.0f);
  float v = 0.0f;
  for (int k = t; k < H; k += 256) v += (sums[g * H + k] / cnt) * W_out[k];
  red[t] = v;
  __syncthreads();
  for (int s = 128; s > 0; s >>= 1) {
    if (t < s) red[t] += red[t + s];
    __syncthreads();
  }
  if (t == 0) out[g] = fmaxf(red[0] + b_out[0], 0.0f);
}

// ---------------------------------------------------------------------------
extern "C" void kernel_launch(void* const* d_in, const int* in_sizes, int n_in,
                              void* d_out, int out_size, void* d_ws,
                              size_t ws_size, hipStream_t stream) {
  const float* x = (const float*)d_in[0];
  const int* ei = (const int*)d_in[1];     // [2,E]: src row then dst row
  const int* batch = (const int*)d_in[2];  // [N], sorted
  const float* W_l = (const float*)d_in[3];
  const float* b_l = (const float*)d_in[4];
  const float* W_out = (const float*)d_in[5];
  const float* b_out = (const float*)d_in[6];
  float* out = (float*)d_out;

  const int N = in_sizes[0];
  const int E = in_sizes[1] / 2;
  const int H = in_sizes[3];
  const int G = out_size;

  float* agg = (float*)d_ws;             // N
  float* sums = agg + N;                 // G*H
  float* counts = sums + (size_t)G * H;  // G

  const int total = N + G * H + G;
  zero_ws_kernel<<<(total + 255) / 256, 256, 0, stream>>>((float*)d_ws, total);

  if (E > 0 && (E & 3) == 0) {
    const int E4 = E >> 2;
    edge_scatter_vec4_kernel<<<(E4 + 255) / 256, 256, 0, stream>>>(
        (const v4i*)ei, (const v4i*)(ei + E), x, agg, E4);
  } else if (E > 0) {
    int eb = (E + 255) / 256;
    if (eb > 16384) eb = 16384;
    edge_scatter_scalar_kernel<<<eb, 256, 0, stream>>>(ei, ei + E, x, agg, E);
  }

  node_pool_wmma_kernel<<<(N + NPB - 1) / NPB, 256, 0, stream>>>(
      agg, batch, W_l, b_l, sums, counts, N, H, G);

  finalize_kernel<<<G, 256, 0, stream>>>(sums, counts, W_out, b_out, out, H);
}